// CoupledFUSERoute_1803886265489
// MI455X (gfx1250) — compile-verified
//
#include <hip/hip_runtime.h>
#include <math.h>

// ---------------------------------------------------------------------------
// Muskingum-Cunge tree routing as a cross-WGP software pipeline (CDNA5).
//   - 14 tree levels -> pipeline stages, one or more 256-thread blocks each
//     (level 0 split into 16 chunks, level 1 into 8, ...; 40 blocks total).
//   - Per-reach state (Q, I_prev) and hydraulic constants live in registers
//     for the entire 2048-step scan (zero state traffic).
//   - Level handshakes: agent-scope produce/consume counters + depth-2 ring
//     of q buffers in d_ws.
//   - Lateral inflows: double-buffered in LDS via CDNA5 async-to-LDS DMA
//     (global_load_async_to_lds_b32, ASYNCcnt) issued one timestep ahead so
//     the DMA hides behind the handshake spin and substep math.
// ---------------------------------------------------------------------------

#define N_LEVELSC   14
#define N_REACHESC  16383
#define QSTRIDE     16384         // padded ring stride (floats) -> 8B-aligned slots
#define T_STEPS     2048
#define DT_SUBF     21600.0f      // 86400 / 4
#define EPS_F       1e-6f

#define CHUNK       512           // reaches per block
#define NTHREADS    256
#define MAXK        2             // CHUNK / NTHREADS
#define NBLOCKS     40            // 16+8+4+2 + 10*1
#define FLAG_STRIDE 16            // 64B between counters
#define FLAG_INTS   ((2 * N_LEVELSC) * FLAG_STRIDE)

__constant__ int c_lvl_size[N_LEVELSC] = {8192,4096,2048,1024,512,256,128,64,32,16,8,4,2,1};
__constant__ int c_lvl_off [N_LEVELSC] = {0,8192,12288,14336,15360,15872,16128,16256,16320,16352,16368,16376,16380,16382};
__constant__ int c_lvl_nblk[N_LEVELSC] = {16,8,4,2,1,1,1,1,1,1,1,1,1,1};

// ---- CDNA5 async global->LDS (ASYNCcnt tracked, CDNA5 ISA) -------------
__device__ __forceinline__ void async_load_f32_to_lds(const float* gsrc, float* lds_generic) {
  unsigned           lds_off = (unsigned)(unsigned long long)(uintptr_t)lds_generic; // low 32b = LDS byte offset
  unsigned long long ga      = (unsigned long long)(uintptr_t)gsrc;
  asm volatile("global_load_async_to_lds_b32 %0, %1, off"
               :: "v"(lds_off), "v"(ga) : "memory");
}
__device__ __forceinline__ void wait_asynccnt0() {
  asm volatile("s_wait_asynccnt 0x0" ::: "memory");
}

// ---- agent-scope spin ------------------------------------------------------
__device__ __forceinline__ void spin_ge(int* p, int target) {
  while (__hip_atomic_load(p, __ATOMIC_ACQUIRE, __HIP_MEMORY_SCOPE_AGENT) < target) {
    __builtin_amdgcn_s_sleep(1);
  }
}

// ---- one Muskingum-Cunge substep ------------------------------------------
__device__ __forceinline__ float mc_substep(float In, float Io, float q,
                                            float dx, float sqSn, float i2Sdx,
                                            float wc, float we, float dc, float de)
{
  float Qref  = fmaxf((In + Io + q) * (1.0f / 3.0f), EPS_F);
  float width = fmaxf(wc * powf(Qref, we), EPS_F);
  float depth = fmaxf(dc * powf(Qref, de), EPS_F);
  float v     = powf(depth, 0.6666666667f) * sqSn;
  float c     = fmaxf(1.6666666667f * v, EPS_F);
  float K     = dx / c;
  float X     = 0.5f - (Qref * i2Sdx) / (c * width);
  X = fminf(fmaxf(X, 0.0f), 0.5f);
  float a2KX  = 2.0f * K * X;
  float a2K1X = 2.0f * K * (1.0f - X);
  float D     = a2K1X + DT_SUBF;
  float invD  = 1.0f / D;
  float C0 = (DT_SUBF - a2KX) * invD;
  float C1 = (DT_SUBF + a2KX) * invD;
  float C2 = (a2K1X - DT_SUBF) * invD;
  return fmaxf(C0 * In + C1 * Io + C2 * q, 0.0f);
}

__global__ void __launch_bounds__(NTHREADS, 1) init_flags_k(int* flags) {
  for (int i = threadIdx.x; i < FLAG_INTS; i += NTHREADS) flags[i] = 0;
}

__global__ void __launch_bounds__(NTHREADS, 1)
mc_route_kernel(const float* __restrict__ lat,
                const float* __restrict__ log_n,
                const float* __restrict__ lengths,
                const float* __restrict__ slopes,
                const float* __restrict__ wcoef,
                const float* __restrict__ wexp,
                const float* __restrict__ dcoef,
                const float* __restrict__ dexp,
                float* __restrict__ out,
                float* __restrict__ qbuf,     // 2 * QSTRIDE floats (ring)
                int*   __restrict__ flags)    // prod[14], cons[14] (strided)
{
  const int tid = threadIdx.x;

  // Decode (level, chunk) from flat block id.
  int bid = blockIdx.x;
  int level = N_LEVELSC - 1, chunk = 0;
  {
    int acc = 0;
    for (int l = 0; l < N_LEVELSC; ++l) {
      int nb = c_lvl_nblk[l];
      if (bid < acc + nb) { level = l; chunk = bid - acc; break; }
      acc += nb;
    }
  }
  const int lsize      = c_lvl_size[level];
  const int loff       = c_lvl_off[level];
  const int coff       = chunk * CHUNK;
  const int nloc       = min(lsize - coff, CHUNK);
  const int child_off  = (level > 0) ? c_lvl_off[level - 1] : 0;
  const int nblk_child = (level > 0) ? c_lvl_nblk[level - 1] : 0;
  const int nblk_cons  = (level < N_LEVELSC - 1) ? c_lvl_nblk[level + 1] : 0;

  int* prod  = flags;                           // prod[l]  at flags[l*FLAG_STRIDE]
  int* consf = flags + N_LEVELSC * FLAG_STRIDE; // cons[l]  at consf[l*FLAG_STRIDE]

  __shared__ float lat_lds[2][CHUNK];

  bool  act[MAXK];
  int   loc[MAXK], g[MAXK];
  float dx[MAXK], sqSn[MAXK], i2Sdx[MAXK];
  float wc_[MAXK], we_[MAXK], dc_[MAXK], de_[MAXK];
  float Qs[MAXK], Ip[MAXK];

#pragma unroll
  for (int k = 0; k < MAXK; ++k) {
    int li = k * NTHREADS + tid;
    act[k] = li < nloc;
    loc[k] = coff + li;
    g[k]   = loff + loc[k];
    Qs[k] = 0.0f; Ip[k] = 0.0f;
    dx[k] = 1.0f; sqSn[k] = 1.0f; i2Sdx[k] = 1.0f;
    wc_[k] = 1.0f; we_[k] = 0.5f; dc_[k] = 1.0f; de_[k] = 0.5f;
    if (act[k]) {
      float n  = expf(log_n[g[k]]);
      float S  = slopes[g[k]];
      float L  = lengths[g[k]];
      dx[k]    = L;
      sqSn[k]  = sqrtf(S) / n;
      i2Sdx[k] = 1.0f / (2.0f * S * L);
      wc_[k] = wcoef[g[k]]; we_[k] = wexp[g[k]];
      dc_[k] = dcoef[g[k]]; de_[k] = dexp[g[k]];
    }
  }

  // Prime async prefetch of the t=0 lateral slice into LDS slot 0.
#pragma unroll
  for (int k = 0; k < MAXK; ++k)
    if (act[k]) async_load_f32_to_lds(lat + g[k], &lat_lds[0][k * NTHREADS + tid]);

  for (int t = 0; t < T_STEPS; ++t) {
    const int slot = t & 1;

    // (a) Cross-level handshakes: wait for child data of step t, and for our
    //     ring slot (holding data of t-2) to be fully consumed.
    if (tid == 0) {
      if (level > 0)               spin_ge(&prod[(level - 1) * FLAG_STRIDE], nblk_child * (t + 1));
      if (nblk_cons > 0 && t >= 2) spin_ge(&consf[level * FLAG_STRIDE],      nblk_cons  * (t - 1));
    }
    __syncthreads();
    if (level > 0) __threadfence();   // acquire: see child's agent-scope q stores

    // (b) Lateral slice for this timestep has landed in our LDS slot
    //     (own-wave ASYNCcnt suffices: each lane reads slots it issued).
    wait_asynccnt0();
    float inflow[MAXK];
#pragma unroll
    for (int k = 0; k < MAXK; ++k)
      inflow[k] = act[k] ? lat_lds[slot][k * NTHREADS + tid] : 0.0f;

    // (c) Immediately kick off async DMA of the NEXT timestep's lateral slice
    //     (other LDS slot) so it overlaps child loads, substep math and the
    //     next handshake spin.
    if (t + 1 < T_STEPS) {
      const float* row = lat + (size_t)(t + 1) * N_REACHESC;
#pragma unroll
      for (int k = 0; k < MAXK; ++k)
        if (act[k]) async_load_f32_to_lds(row + g[k], &lat_lds[slot ^ 1][k * NTHREADS + tid]);
    }

    // (d) Add routed upstream flow: children (2j, 2j+1) -> parent j.
    //     QSTRIDE even + child_off even + 2*loc even => 8B-aligned float2.
    if (level > 0) {
#pragma unroll
      for (int k = 0; k < MAXK; ++k) {
        if (act[k]) {
          const float2 cq = *(const float2*)(qbuf + (size_t)slot * QSTRIDE + child_off + 2 * loc[k]);
          inflow[k] += cq.x + cq.y;
        }
      }
    }

    // (e) 4 Muskingum-Cunge substeps; publish q into our ring slot.
#pragma unroll
    for (int k = 0; k < MAXK; ++k) {
      if (!act[k]) continue;
      float q = Qs[k], io = Ip[k];
      const float in = inflow[k];
#pragma unroll
      for (int s = 0; s < 4; ++s) {
        q  = mc_substep(in, io, q, dx[k], sqSn[k], i2Sdx[k], wc_[k], we_[k], dc_[k], de_[k]);
        io = in;
      }
      Qs[k] = q; Ip[k] = in;
      qbuf[(size_t)slot * QSTRIDE + g[k]] = q;
    }
    if (level == N_LEVELSC - 1 && tid == 0) out[t] = Qs[0];   // outlet discharge

    // (f) Release: push q to agent scope (only if someone consumes it),
    //     then bump flags.
    if (nblk_cons > 0) __threadfence();
    __syncthreads();
    if (tid == 0) {
      if (level > 0)
        __hip_atomic_fetch_add(&consf[(level - 1) * FLAG_STRIDE], 1,
                               __ATOMIC_RELEASE, __HIP_MEMORY_SCOPE_AGENT);
      if (nblk_cons > 0)
        __hip_atomic_fetch_add(&prod[level * FLAG_STRIDE], 1,
                               __ATOMIC_RELEASE, __HIP_MEMORY_SCOPE_AGENT);
    }
  }
}

extern "C" void kernel_launch(void* const* d_in, const int* in_sizes, int n_in,
                              void* d_out, int out_size, void* d_ws, size_t ws_size,
                              hipStream_t stream) {
  (void)in_sizes; (void)n_in; (void)out_size; (void)ws_size;
  const float* lat   = (const float*)d_in[0];
  const float* log_n = (const float*)d_in[1];
  const float* len   = (const float*)d_in[2];
  const float* slo   = (const float*)d_in[3];
  const float* wc    = (const float*)d_in[4];
  const float* we    = (const float*)d_in[5];
  const float* dc    = (const float*)d_in[6];
  const float* de    = (const float*)d_in[7];

  // Workspace layout: [ qbuf: 2*QSTRIDE floats ][ flags: 448 ints ]  (~133 KB)
  float* qbuf  = (float*)d_ws;
  int*   flags = (int*)((char*)d_ws + (size_t)2 * QSTRIDE * sizeof(float));

  init_flags_k<<<1, NTHREADS, 0, stream>>>(flags);
  mc_route_kernel<<<NBLOCKS, NTHREADS, 0, stream>>>(
      lat, log_n, len, slo, wc, we, dc, de, (float*)d_out, qbuf, flags);
}